// DetectionCriterion_89558658056211
// MI455X (gfx1250) — compile-verified
//
#include <hip/hip_runtime.h>
#include <stdint.h>

#define BATCH   64
#define NPRED   300
#define NTGT    30
#define THREADS 256
#define BIGF    1e18f

// ---------------------------------------------------------------------------
// GIoU of two xyxy boxes (matches reference giou_pair / giou_matrix math)
// ---------------------------------------------------------------------------
__device__ __forceinline__ float giou_xyxy(float ax1, float ay1, float ax2, float ay2,
                                           float bx1, float by1, float bx2, float by2) {
    float area_a = (ax2 - ax1) * (ay2 - ay1);
    float area_b = (bx2 - bx1) * (by2 - by1);
    float ltx = fmaxf(ax1, bx1), lty = fmaxf(ay1, by1);
    float rbx = fminf(ax2, bx2), rby = fminf(ay2, by2);
    float wx = fmaxf(rbx - ltx, 0.0f), wy = fmaxf(rby - lty, 0.0f);
    float inter = wx * wy;
    float uni = area_a + area_b - inter;
    float iou = inter / uni;
    float cx1 = fminf(ax1, bx1), cy1 = fminf(ay1, by1);
    float cx2 = fmaxf(ax2, bx2), cy2 = fmaxf(ay2, by2);
    float cw = fmaxf(cx2 - cx1, 0.0f), ch = fmaxf(cy2 - cy1, 0.0f);
    float areac = cw * ch;
    return iou - (areac - uni) / areac;
}

#if defined(__HIP_DEVICE_COMPILE__) && __has_builtin(__builtin_amdgcn_global_load_async_to_lds_b32)
#define USE_ASYNC_LDS 1
typedef __attribute__((address_space(1))) int g1_int;   // global (AS1) int
typedef __attribute__((address_space(3))) int l3_int;   // LDS (AS3) int
__device__ __forceinline__ void async_copy_b32(const float* g, float* l) {
    __builtin_amdgcn_global_load_async_to_lds_b32(
        (g1_int*)(uintptr_t)g,
        (l3_int*)(uint32_t)(uintptr_t)l, 0, 0);
}
__device__ __forceinline__ void wait_async0() {
#if __has_builtin(__builtin_amdgcn_s_wait_asynccnt)
    __builtin_amdgcn_s_wait_asynccnt(0);
#else
    asm volatile("s_wait_asynccnt 0" ::: "memory");
#endif
}
#endif

// ---------------------------------------------------------------------------
// One block per batch element: cost matrix -> Hungarian (wave 0) -> losses
// ---------------------------------------------------------------------------
__global__ __launch_bounds__(THREADS)
void detr_criterion_kernel(const float* __restrict__ pred_boxes,   // [B,N,4] xywh
                           const float* __restrict__ pred_classes, // [B,N]
                           const float* __restrict__ targets,      // [B,T,4] xywh
                           float* __restrict__ ws_partial) {       // [B,3]
    __shared__ float costL[NTGT * NPRED];   // cost[t][n] (transposed layout for LSA)
    __shared__ float pbL[NPRED * 4];
    __shared__ float tgL[NTGT * 4];
    __shared__ float pcL[NPRED];
    __shared__ float vL[NPRED];
    __shared__ float spcL[NPRED];
    __shared__ int   pathL[NPRED];
    __shared__ int   scL[NPRED];
    __shared__ int   row4colL[NPRED];
    __shared__ float uL[NTGT];
    __shared__ int   srL[NTGT];
    __shared__ int   col4rowL[NTGT];
    __shared__ int   sortedL[NTGT];
    __shared__ int   flagL[NPRED];
    __shared__ float redL[THREADS * 3];

    const int b   = blockIdx.x;
    const int tid = threadIdx.x;
    const float* pb = pred_boxes   + (size_t)b * NPRED * 4;
    const float* pc = pred_classes + (size_t)b * NPRED;
    const float* tg = targets      + (size_t)b * NTGT * 4;

    // ---------------- Phase 0: stage inputs into LDS --------------------
#ifdef USE_ASYNC_LDS
    for (int i = tid; i < NPRED * 4; i += THREADS) async_copy_b32(pb + i, &pbL[i]);
    for (int i = tid; i < NPRED;     i += THREADS) async_copy_b32(pc + i, &pcL[i]);
    for (int i = tid; i < NTGT * 4;  i += THREADS) async_copy_b32(tg + i, &tgL[i]);
    wait_async0();
#else
    __builtin_prefetch(pb + tid, 0, 3);
    for (int i = tid; i < NPRED * 4; i += THREADS) pbL[i] = pb[i];
    for (int i = tid; i < NPRED;     i += THREADS) pcL[i] = pc[i];
    for (int i = tid; i < NTGT * 4;  i += THREADS) tgL[i] = tg[i];
#endif
    __syncthreads();

    // ---------------- Phase 1: cost[t][n] + state init ------------------
    for (int idx = tid; idx < NTGT * NPRED; idx += THREADS) {
        int t = idx / NPRED;
        int n = idx - t * NPRED;
        float px = pbL[n * 4 + 0], py = pbL[n * 4 + 1];
        float pw = pbL[n * 4 + 2], ph = pbL[n * 4 + 3];
        float tx = tgL[t * 4 + 0], ty = tgL[t * 4 + 1];
        float tw = tgL[t * 4 + 2], th = tgL[t * 4 + 3];
        float l1 = fabsf(px - tx) + fabsf(py - ty) + fabsf(pw - tw) + fabsf(ph - th);
        float g  = giou_xyxy(px, py, px + pw, py + ph, tx, ty, tx + tw, ty + th);
        costL[t * NPRED + n] = -pcL[n] - g + l1;
    }
    for (int c = tid; c < NPRED; c += THREADS) {
        row4colL[c] = -1; vL[c] = 0.0f; flagL[c] = 0;
    }
    if (tid < NTGT) { col4rowL[tid] = -1; uL[tid] = 0.0f; }
    __syncthreads();

    // ---------------- Phase 2: Hungarian on wave 0 ----------------------
    if (tid < 32) {
        const int lane = tid;
        for (int row = 0; row < NTGT; ++row) {
            if (lane < NTGT) srL[lane] = 0;
            for (int c = lane; c < NPRED; c += 32) {
                scL[c] = 0; spcL[c] = BIGF; pathL[c] = 0;
            }
            int   curI   = row;
            int   sink   = -1;
            float minVal = 0.0f;

            while (sink < 0) {
                srL[curI] = 1;                      // uniform value, all lanes
                float ucur = uL[curI];
                float bv = __builtin_inff();
                int   bi = NPRED;
                for (int c = lane; c < NPRED; c += 32) {
                    if (!scL[c]) {
                        float red = minVal + costL[curI * NPRED + c] - ucur - vL[c];
                        float s = spcL[c];
                        if (red < s) { spcL[c] = red; pathL[c] = curI; s = red; }
                        if (s < bv || (s == bv && c < bi)) { bv = s; bi = c; }
                    }
                }
                // wave32 butterfly argmin, first-index tie-break (== jnp.argmin)
#pragma unroll
                for (int off = 16; off >= 1; off >>= 1) {
                    float ov = __shfl_xor(bv, off, 32);
                    int   oi = __shfl_xor(bi, off, 32);
                    if (ov < bv || (ov == bv && oi < bi)) { bv = ov; bi = oi; }
                }
                minVal = bv;
                int j = bi;
                scL[j] = 1;                         // uniform value, all lanes
                int r = row4colL[j];                // broadcast read
                if (r < 0) sink = j; else curI = r;
            }

            // dual updates (before augmentation, like the reference)
            if (lane < NTGT && srL[lane]) {
                if (lane == row) uL[lane] += minVal;
                else             uL[lane] += minVal - spcL[col4rowL[lane]];
            }
            for (int c = lane; c < NPRED; c += 32)
                if (scL[c]) vL[c] -= (minVal - spcL[c]);

            // augment alternating path back from sink (uniform across lanes)
            int jj = sink;
            while (true) {
                int i = pathL[jj];
                row4colL[jj] = i;
                int nj = col4rowL[i];
                col4rowL[i] = jj;
                if (i == row) break;
                jj = nj;
            }
        }

        // boxes_idx = sort(col4row); values are distinct -> counting rank
        if (lane < NTGT) {
            int vme  = col4rowL[lane];
            int rank = 0;
            for (int j = 0; j < NTGT; ++j) rank += (col4rowL[j] < vme) ? 1 : 0;
            sortedL[rank] = vme;
            flagL[vme] = 1;                         // matched-query label
        }
    }
    __syncthreads();

    // ---------------- Phase 3: losses (partials per batch) --------------
    float a_cls = 0.0f, a_bbox = 0.0f, a_giou = 0.0f;
    for (int n = tid; n < NPRED; n += THREADS) {
        float p   = pcL[n];
        float lab = (float)flagL[n];
        float lg  = fmaxf(logf(p),        -100.0f);
        float lg1 = fmaxf(logf(1.0f - p), -100.0f);
        a_cls += -(lab * lg + (1.0f - lab) * lg1);
    }
    if (tid < NTGT) {
        int s = sortedL[tid];
        float px = pbL[s * 4 + 0], py = pbL[s * 4 + 1];
        float pw = pbL[s * 4 + 2], ph = pbL[s * 4 + 3];
        float tx = tgL[tid * 4 + 0], ty = tgL[tid * 4 + 1];
        float tw = tgL[tid * 4 + 2], th = tgL[tid * 4 + 3];
        a_bbox += fabsf(px - tx) + fabsf(py - ty) + fabsf(pw - tw) + fabsf(ph - th);
        float g = giou_xyxy(px, py, px + pw, py + ph, tx, ty, tx + tw, ty + th);
        a_giou += 1.0f - g;
    }
    redL[tid]               = a_cls;
    redL[THREADS + tid]     = a_bbox;
    redL[2 * THREADS + tid] = a_giou;
    __syncthreads();
    for (int s = THREADS / 2; s > 0; s >>= 1) {
        if (tid < s) {
            redL[tid]               += redL[tid + s];
            redL[THREADS + tid]     += redL[THREADS + tid + s];
            redL[2 * THREADS + tid] += redL[2 * THREADS + tid + s];
        }
        __syncthreads();
    }
    if (tid == 0) {
        float* w = ws_partial + (size_t)b * 3;
        w[0] = redL[0];
        w[1] = redL[THREADS];
        w[2] = redL[2 * THREADS];
    }
}

// ---------------------------------------------------------------------------
// Deterministic fixed-order final reduction over the 64 batch partials
// ---------------------------------------------------------------------------
__global__ void detr_finalize_kernel(const float* __restrict__ ws, float* __restrict__ out) {
    if (blockIdx.x == 0 && threadIdx.x == 0) {
        float c = 0.0f, bb = 0.0f, gg = 0.0f;
        for (int b = 0; b < BATCH; ++b) {
            c  += ws[b * 3 + 0];
            bb += ws[b * 3 + 1];
            gg += ws[b * 3 + 2];
        }
        out[0] = c  / (float)(BATCH * NPRED);  // bce.mean()
        out[1] = bb / (float)(BATCH * NTGT);   // / num_boxes
        out[2] = gg / (float)(BATCH * NTGT);   // / num_boxes
    }
}

extern "C" void kernel_launch(void* const* d_in, const int* in_sizes, int n_in,
                              void* d_out, int out_size, void* d_ws, size_t ws_size,
                              hipStream_t stream) {
    (void)in_sizes; (void)n_in; (void)out_size; (void)ws_size;
    const float* pred_boxes   = (const float*)d_in[0];
    const float* pred_classes = (const float*)d_in[1];
    const float* targets      = (const float*)d_in[2];
    float* ws  = (float*)d_ws;
    float* out = (float*)d_out;

    detr_criterion_kernel<<<BATCH, THREADS, 0, stream>>>(pred_boxes, pred_classes,
                                                         targets, ws);
    detr_finalize_kernel<<<1, 32, 0, stream>>>(ws, out);
}